// HstackDiag_21225728377282
// MI455X (gfx1250) — compile-verified
//
#include <hip/hip_runtime.h>

typedef __attribute__((ext_vector_type(2))) float v2f;
typedef __attribute__((ext_vector_type(8))) float v8f;

#define LDA 68   // padded LDS row stride (floats) for A  (64 + 4)
#define LDW 68   // padded LDS row stride (floats) for W^T
#define LDD 100  // padded LDS row stride (floats) for staged D (92 + 8)

// One workgroup per n. 256 threads = 8 waves (wave32).
// GEMM per n: D[M=128][N=46] = A[M=128][K=64] * W[K=64][N=46]
//   M rows   m  = b*2 + i            (b in [0,64), i in {0,1})
//   K index  kk = h*32 + k*16 + u    (h,k in {0,1}, u in [0,16))
//   N cols   c  = j*23 + t           (j in {0,1}, t in [0,23))
//   W[kk][c] = diag_h[n,k,j,t-u] when 0 <= t-u < 8, else 0
__global__ __launch_bounds__(256) void hstack_diag_wmma(
    const float* __restrict__ x,    // (64, 2*size, 2, 2, 16) fp32
    const float* __restrict__ dg1,  // (size, 2, 2, 8) fp32
    const float* __restrict__ dg2,  // (size, 2, 2, 8) fp32
    float* __restrict__ out,        // (64, size, 2, 2, 23) fp32
    int size)
{
    __shared__ float sA[128 * LDA];   // 34816 B ; reused as sD (64*LDD=6400 floats) in epilogue
    __shared__ float sW[48 * LDW];    // 13056 B  (W stored transposed: sW[c][kk])

    const int n   = blockIdx.x;
    const int tid = threadIdx.x;

    // ---- zero W^T (band matrix: everything not scattered below stays 0) ----
    for (int idx = tid; idx < 48 * LDW; idx += 256) sW[idx] = 0.0f;

    // ---- stage A into LDS: 128 chunks of 64 contiguous floats, float4 loads ----
    {
        const int grp  = tid >> 4;   // 16 groups of 16 lanes
        const int l16g = tid & 15;
        #pragma unroll
        for (int it = 0; it < 8; ++it) {
            int chunk = it * 16 + grp;        // chunk = h*64 + b
            int h = chunk >> 6;
            int b = chunk & 63;
            long gbase = ((long)b * (2L * (long)size) + (long)h * size + n) * 64L;
            float4 val = reinterpret_cast<const float4*>(x + gbase)[l16g];
            int f = l16g * 4;                 // flat (i,k,u) within chunk
            int i = f >> 5;
            int c = f & 31;                   // k*16 + u
            *reinterpret_cast<float4*>(&sA[(b * 2 + i) * LDA + h * 32 + c]) = val;
        }
    }
    __syncthreads();

    // ---- scatter-fill W^T from the 64 diag coefficients of this n ----
    {
        int dv = tid >> 2;           // coefficient id 0..63 = h*32 + k*16 + j*8 + v
        int ug = tid & 3;            // 4 threads share one coefficient
        int h = dv >> 5;
        int r = dv & 31;             // k*16 + j*8 + v (contiguous layout of diag per n)
        int k = r >> 4;
        int j = (r >> 3) & 1;
        int v = r & 7;
        float val = (h ? dg2 : dg1)[(long)n * 32 + r];
        #pragma unroll
        for (int s = 0; s < 4; ++s) {
            int u = ug * 4 + s;      // replicate along the band: t = u + v
            sW[(j * 23 + u + v) * LDW + h * 32 + k * 16 + u] = val;
        }
    }
    __syncthreads();

    // ---- WMMA K-loop: wave w owns M-tile w, three N-tiles ----
    const int wave = tid >> 5;
    const int L    = tid & 31;
    const int l16  = L & 15;         // M (for A) / N (for B) within tile
    const int lh   = L >> 4;         // selects K pair {0,1} vs {2,3}

    v8f acc0 = {}, acc1 = {}, acc2 = {};
    const float* Arow = &sA[(wave * 16 + l16) * LDA];
    const float* W0   = &sW[(l16)      * LDW];
    const float* W1   = &sW[(16 + l16) * LDW];
    const float* W2   = &sW[(32 + l16) * LDW];

    #pragma unroll
    for (int ks = 0; ks < 16; ++ks) {
        int col = ks * 4 + lh * 2;   // K offset for this lane (VGPR pair = K, K+1)
        v2f a  = *reinterpret_cast<const v2f*>(Arow + col);
        v2f b0 = *reinterpret_cast<const v2f*>(W0 + col);
        v2f b1 = *reinterpret_cast<const v2f*>(W1 + col);
        v2f b2 = *reinterpret_cast<const v2f*>(W2 + col);
        acc0 = __builtin_amdgcn_wmma_f32_16x16x4_f32(false, a, false, b0, (short)0, acc0, false, false);
        acc1 = __builtin_amdgcn_wmma_f32_16x16x4_f32(false, a, false, b1, (short)0, acc1, false, false);
        acc2 = __builtin_amdgcn_wmma_f32_16x16x4_f32(false, a, false, b2, (short)0, acc2, false, false);
    }

    // ---- stage D into LDS (reuse sA region; A is dead after this barrier) ----
    __syncthreads();
    float* sD = sA;                  // sD[b][cp], cp = i*46 + j*23 + t, stride LDD
    #pragma unroll
    for (int nt = 0; nt < 3; ++nt) {
        v8f acc = (nt == 0) ? acc0 : (nt == 1) ? acc1 : acc2;
        int col = nt * 16 + l16;
        if (col < 46) {
            #pragma unroll
            for (int vv = 0; vv < 8; ++vv) {
                int rowM = wave * 16 + vv + 8 * lh;   // m = b*2 + i
                int b = rowM >> 1, i = rowM & 1;
                sD[b * LDD + i * 46 + col] = acc[vv];
            }
        }
    }
    __syncthreads();

    // ---- coalesced writeout: 64 b-rows * 23 float4 chunks, b128 stores ----
    {
        float4* out4 = reinterpret_cast<float4*>(out);
        for (int q = tid; q < 64 * 23; q += 256) {
            int b  = q / 23;
            int f4 = q - b * 23;
            float4 val = *reinterpret_cast<const float4*>(&sD[b * LDD + f4 * 4]);
            out4[((long)b * size + n) * 23 + f4] = val;
        }
    }
}

extern "C" void kernel_launch(void* const* d_in, const int* in_sizes, int n_in,
                              void* d_out, int out_size, void* d_ws, size_t ws_size,
                              hipStream_t stream) {
    const float* x   = (const float*)d_in[0];
    const float* dg1 = (const float*)d_in[1];
    const float* dg2 = (const float*)d_in[2];
    float* out = (float*)d_out;
    const int size = in_sizes[1] / 32;   // diag1 has size*2*2*8 elements
    hstack_diag_wmma<<<dim3(size), dim3(256), 0, stream>>>(x, dg1, dg2, out, size);
}